// GraphTransformerAutoencoder_32066225832382
// MI455X (gfx1250) — compile-verified
//
#include <hip/hip_runtime.h>
#include <hip/hip_bf16.h>
#include <math.h>

typedef __attribute__((ext_vector_type(2))) float v2f;
typedef __attribute__((ext_vector_type(8))) float v8f;

#define NN   50000
#define EE   800000
#define HH   2
#define DD   64
#define HD   128      // H*D == IN == 128
#define NM   (NN * HD)
#define MT   5        // M-tiles (of 16 rows) per block: 50000 = 625 * 80

// ---------------------------------------------------------------------------
// GEMM: Out[N,128] = A[N,128] @ W[128,128] + bias[128]
// grid.x = N/80 blocks; block = 256 threads = 8 wave32s.
// Wave w owns cols [w*16, w*16+16) and 5 row tiles (80 rows); B fragments are
// loaded once per K-step and reused across the 5 accumulators (5x less W
// traffic than a 16-row block).
// Uses V_WMMA_F32_16X16X4_F32, K split into 32 steps of 4.
// ISA layouts (CDNA5 ISA):
//   A 16x4 : lane m=(l&15) ; VGPR0 = K=(l>>4)*2, VGPR1 = K+1
//   B 4x16 : lane n=(l&15) ; VGPR0 = K=(l>>4)*2, VGPR1 = K+1
//   C/D    : VGPR j : lanes 0-15 -> row j, lanes 16-31 -> row j+8
// ---------------------------------------------------------------------------
__global__ __launch_bounds__(256) void gemm80(const float* __restrict__ A,
                                              const float* __restrict__ W,
                                              const float* __restrict__ bias,
                                              float* __restrict__ Out) {
  __shared__ float As[MT * 16 * 132];       // 80 rows, padded stride 132 (~42 KB)
  const int row0 = blockIdx.x * (MT * 16);

  // cooperative load of the 80x128 A tile (2560 float4s / 256 threads)
  #pragma unroll
  for (int t = 0; t < 2 * MT; ++t) {
    int i  = threadIdx.x + t * 256;         // float4 index
    int r  = i >> 5;                        // 32 float4 per row
    int c4 = (i & 31) << 2;
    *(float4*)&As[r * 132 + c4] =
        *(const float4*)(A + (size_t)(row0 + r) * HD + c4);
  }
  __syncthreads();

  const int lane = threadIdx.x & 31;
  const int wave = threadIdx.x >> 5;
  const int n0   = wave * 16;
  const int nl   = lane & 15;
  const int hi   = lane >> 4;               // 0 or 1
  const int bk   = hi * 2;                  // K offset within group of 4

  v8f acc[MT];
  #pragma unroll
  for (int m = 0; m < MT; ++m) acc[m] = (v8f){};

  #pragma unroll 4
  for (int kk = 0; kk < 128; kk += 4) {
    const float* wp = W + (size_t)(kk + bk) * HD + n0 + nl;
    v2f b;
    b.x = wp[0];
    b.y = wp[HD];
    #pragma unroll
    for (int m = 0; m < MT; ++m) {
      v2f a = *(const v2f*)&As[(m * 16 + nl) * 132 + kk + bk];
      acc[m] = __builtin_amdgcn_wmma_f32_16x16x4_f32(
          /*neg_a=*/false, a, /*neg_b=*/false, b,
          /*c_mod=*/(short)0, acc[m], /*reuse_a=*/false, /*reuse_b=*/false);
    }
  }

  const float bv = bias[n0 + nl];
  #pragma unroll
  for (int m = 0; m < MT; ++m) {
    float* orow = Out + (size_t)(row0 + m * 16 + hi * 8) * HD + n0 + nl;
    #pragma unroll
    for (int j = 0; j < 8; ++j)
      orow[(size_t)j * HD] = acc[m][j] + bv;
  }
}

// ---------------------------------------------------------------------------
__global__ __launch_bounds__(256) void zero_f4(float4* __restrict__ p, int n4) {
  int i = blockIdx.x * 256 + threadIdx.x;
  if (i < n4) p[i] = make_float4(0.f, 0.f, 0.f, 0.f);
}

// monotonic uint key for f32 atomic max
__device__ __forceinline__ unsigned fkey(float f) {
  unsigned b = __float_as_uint(f);
  return (b & 0x80000000u) ? ~b : (b | 0x80000000u);
}

// per-(edge,head): score = q[dst].k[src]/8 ; atomic segment-max into mkey
__global__ __launch_bounds__(256) void edge_score_max(
    const int* __restrict__ ei, const float* __restrict__ q,
    const float* __restrict__ k, float* __restrict__ score,
    unsigned* __restrict__ mkey) {
  int tid = blockIdx.x * 256 + threadIdx.x;
  if (tid >= EE * HH) return;
  int e = tid >> 1, h = tid & 1;
  int src = ei[e], dst = ei[EE + e];
  const float4* qr = (const float4*)(q + (size_t)dst * HD + h * DD);
  const float4* kr = (const float4*)(k + (size_t)src * HD + h * DD);
  float s = 0.f;
  #pragma unroll
  for (int i = 0; i < 16; ++i) {
    float4 a = qr[i], b = kr[i];
    s = fmaf(a.x, b.x, s); s = fmaf(a.y, b.y, s);
    s = fmaf(a.z, b.z, s); s = fmaf(a.w, b.w, s);
  }
  s *= 0.125f;                     // 1/sqrt(64)
  score[tid] = s;                  // layout [e][h]
  atomicMax(mkey + (size_t)dst * HH + h, fkey(s));
}

// decode mkey -> float max in place; key==0 => no in-edges => m = 0
__global__ __launch_bounds__(256) void decode_max(unsigned* __restrict__ mk) {
  int tid = blockIdx.x * 256 + threadIdx.x;
  if (tid >= NN * HH) return;
  unsigned kb = mk[tid];
  float m = 0.f;
  if (kb != 0u)
    m = (kb & 0x80000000u) ? __uint_as_float(kb & 0x7FFFFFFFu)
                           : __uint_as_float(~kb);
  ((float*)mk)[tid] = m;
}

// per-(edge,head): e = exp(s - m[dst]); denom += e; agg[dst] += v[src]*e
__global__ __launch_bounds__(256) void edge_accum(
    const int* __restrict__ ei, const float* __restrict__ score,
    const float* __restrict__ m, const float* __restrict__ v,
    float* __restrict__ denom, float* __restrict__ agg) {
  int tid = blockIdx.x * 256 + threadIdx.x;
  if (tid >= EE * HH) return;
  int e = tid >> 1, h = tid & 1;
  int src = ei[e], dst = ei[EE + e];
  float ex = expf(score[tid] - m[(size_t)dst * HH + h]);
  atomicAdd(denom + (size_t)dst * HH + h, ex);
  const float4* vr = (const float4*)(v + (size_t)src * HD + h * DD);
  float* ar = agg + (size_t)dst * HD + h * DD;
  #pragma unroll
  for (int i = 0; i < 16; ++i) {
    float4 t = vr[i];
    atomicAdd(ar + i * 4 + 0, t.x * ex);
    atomicAdd(ar + i * 4 + 1, t.y * ex);
    atomicAdd(ar + i * 4 + 2, t.z * ex);
    atomicAdd(ar + i * 4 + 3, t.w * ex);
  }
}

// z = agg/denom + skip   (normalization folded: sum(v*e)/sum(e))
__global__ __launch_bounds__(256) void node_z(
    const float* __restrict__ agg, const float* __restrict__ denom,
    const float* __restrict__ skip, float* __restrict__ z) {
  int tid = blockIdx.x * 256 + threadIdx.x;
  if (tid >= NM) return;
  int n = tid >> 7, c = tid & 127, h = c >> 6;
  float dn = denom[(size_t)n * HH + h] + 1e-16f;
  z[tid] = agg[tid] / dn + skip[tid];
}

// ---------------------------------------------------------------------------
extern "C" void kernel_launch(void* const* d_in, const int* in_sizes, int n_in,
                              void* d_out, int out_size, void* d_ws, size_t ws_size,
                              hipStream_t stream) {
  const float* x     = (const float*)d_in[0];
  const int*   ei    = (const int*)  d_in[1];
  const float* Wq    = (const float*)d_in[2];
  const float* bq    = (const float*)d_in[3];
  const float* Wk    = (const float*)d_in[4];
  const float* bk    = (const float*)d_in[5];
  const float* Wv    = (const float*)d_in[6];
  const float* bv    = (const float*)d_in[7];
  const float* Wsk   = (const float*)d_in[8];
  const float* bsk   = (const float*)d_in[9];
  const float* Wdec  = (const float*)d_in[10];
  const float* bdec  = (const float*)d_in[11];

  float* out  = (float*)d_out;
  float* xhat = out;                       // output 0: [N,128]
  float* z    = out + (size_t)NM;          // output 1: [N,128]

  // workspace layout (floats): q | k | v | skip | agg | mkey | denom | score
  float* ws    = (float*)d_ws;
  float* q     = ws;
  float* kk    = q  + (size_t)NM;
  float* vv    = kk + (size_t)NM;
  float* skip  = vv + (size_t)NM;
  float* agg   = skip + (size_t)NM;
  float* mbuf  = agg  + (size_t)NM;        // mkey, decoded to float in place
  float* denom = mbuf + (size_t)NN * HH;
  float* score = denom + (size_t)NN * HH;

  const int mBlocks = NN / (MT * 16);      // 625 (exact)

  // 1) projections
  gemm80<<<mBlocks, 256, 0, stream>>>(x, Wq,  bq,  q);
  gemm80<<<mBlocks, 256, 0, stream>>>(x, Wk,  bk,  kk);
  gemm80<<<mBlocks, 256, 0, stream>>>(x, Wv,  bv,  vv);
  gemm80<<<mBlocks, 256, 0, stream>>>(x, Wsk, bsk, skip);

  // 2) zero agg + mkey + denom (contiguous: NM + 2*N*H = 6,600,000 floats)
  int zero4 = (NM + 2 * NN * HH) / 4;
  zero_f4<<<(zero4 + 255) / 256, 256, 0, stream>>>((float4*)agg, zero4);

  // 3) attention: scores + segment max
  int eThreads = EE * HH;
  edge_score_max<<<eThreads / 256, 256, 0, stream>>>(ei, q, kk, score,
                                                     (unsigned*)mbuf);
  decode_max<<<(NN * HH + 255) / 256, 256, 0, stream>>>((unsigned*)mbuf);

  // 4) exp / weighted scatter
  edge_accum<<<eThreads / 256, 256, 0, stream>>>(ei, score, mbuf, vv, denom, agg);

  // 5) z = agg/denom + skip
  node_z<<<NM / 256, 256, 0, stream>>>(agg, denom, skip, z);

  // 6) decoder
  gemm80<<<mBlocks, 256, 0, stream>>>(z, Wdec, bdec, xhat);
}